// crop_predict_32177894981928
// MI455X (gfx1250) — compile-verified
//
#include <hip/hip_runtime.h>
#include <hip/hip_bf16.h>
#include <cstdint>

#define AS1 __attribute__((address_space(1)))
#define AS3 __attribute__((address_space(3)))

typedef int v4i __attribute__((vector_size(16)));

constexpr int B = 4, J = 17, D = 64, G = 128;
constexpr int D3 = D * D * D;          // 262144
constexpr size_t G3 = (size_t)G * G * G;

// ---------------------------------------------------------------------------
// CDNA5 async global->LDS copy (16 bytes per lane) + ASYNCcnt wait
// ---------------------------------------------------------------------------
__device__ __forceinline__ void async_copy_b128(const float* g, float* l) {
#if __has_builtin(__builtin_amdgcn_global_load_async_to_lds_b128)
  __builtin_amdgcn_global_load_async_to_lds_b128(
      (AS1 v4i*)(uintptr_t)g, (AS3 v4i*)(AS3 float*)l, 0, 0);
#else
  unsigned lds_off = (unsigned)(uintptr_t)(AS3 float*)l;
  asm volatile("global_load_async_to_lds_b128 %0, %1, off"
               :
               : "v"(lds_off), "v"((unsigned long long)(uintptr_t)g)
               : "memory");
#endif
}

__device__ __forceinline__ void wait_async0() {
#if __has_builtin(__builtin_amdgcn_s_wait_asynccnt)
  __builtin_amdgcn_s_wait_asynccnt(0);
#else
  asm volatile("s_wait_asynccnt 0" ::: "memory");
#endif
}

// ---------------------------------------------------------------------------
// Kernel 3 (defined first so the disasm snippet shows the async DMA path):
// gather. One block per (b,j,gx). Stage source plane hm[b,j,ix[gx],:,:]
// (16 KB) in LDS via CDNA5 async DMA, then emit 128x128 gathered outputs
// with float4 coalesced stores.
// ---------------------------------------------------------------------------
__global__ void k_gather(const float* __restrict__ hm,
                         const int*   __restrict__ idx,
                         float* __restrict__ out) {
  const int blk = blockIdx.x;                 // ((b*J + j)*G + gx)
  const int gx  = blk & (G - 1);
  const int bj  = blk >> 7;                   // b*J + j
  const int b   = bj / J;
  const int tid = threadIdx.x;                // 256 threads

  __shared__ __align__(16) float plane[D * D];   // 16 KB
  __shared__ int s_iy[G], s_iz[G];

  const int* idxb = idx + b * 3 * G;
  if (tid < G) s_iy[tid]     = idxb[G + tid];
  else         s_iz[tid - G] = idxb[2 * G + (tid - G)];

  const int ix = idxb[gx];
  const float* src = hm + (size_t)bj * D3 + (size_t)ix * (D * D);

  // 16 KB plane = 4 rounds of 256 lanes x 16 B async DMA
#pragma unroll
  for (int k = 0; k < 4; ++k) {
    const int e = (tid + k * 256) * 4;        // float index, 16B aligned
    async_copy_b128(src + e, &plane[e]);
  }
  wait_async0();
  __syncthreads();

  float* dst = out + (size_t)bj * G3 + (size_t)gx * (G * G);
  for (int q = tid; q < (G * G) / 4; q += 256) {
    const int gy  = q >> 5;
    const int gz0 = (q & 31) * 4;
    const float* row = &plane[s_iy[gy] * D];
    float4 v;
    v.x = row[s_iz[gz0 + 0]];
    v.y = row[s_iz[gz0 + 1]];
    v.z = row[s_iz[gz0 + 2]];
    v.w = row[s_iz[gz0 + 3]];
    ((float4*)dst)[q] = v;                    // element offset gy*128+gz0 == 4q
  }
}

// ---------------------------------------------------------------------------
// Kernel 1: per-(b,j) max + mean coordinate of argmax ties
//   pos[bj*3 + {0,1,2}] = (sum_x/cnt, sum_y/cnt, sum_z/cnt)
// ---------------------------------------------------------------------------
__global__ void k_argmax_mean(const float* __restrict__ hm,
                              float* __restrict__ pos) {
  const int bj  = blockIdx.x;                 // 0 .. B*J-1
  const int tid = threadIdx.x;                // 256 threads
  const float* p = hm + (size_t)bj * D3;

  float m = -INFINITY, cnt = 0.f, sx = 0.f, sy = 0.f, sz = 0.f;

  for (int q = tid; q < D3 / 4; q += 256) {
    const int lin = q * 4;
    const float4 v = ((const float4*)p)[q];
    const float x  = (float)(lin >> 12);
    const float y  = (float)((lin >> 6) & 63);
    const int   z0 = lin & 63;
    const float vals[4] = {v.x, v.y, v.z, v.w};
#pragma unroll
    for (int k = 0; k < 4; ++k) {
      const float val = vals[k];
      if (val > m)       { m = val; cnt = 1.f; sx = x; sy = y; sz = (float)(z0 + k); }
      else if (val == m) { cnt += 1.f; sx += x; sy += y; sz += (float)(z0 + k); }
    }
  }

  __shared__ float sM[256], sC[256], sX[256], sY[256], sZ[256];
  sM[tid] = m; sC[tid] = cnt; sX[tid] = sx; sY[tid] = sy; sZ[tid] = sz;
  __syncthreads();

  for (int s = 128; s > 0; s >>= 1) {
    if (tid < s) {
      const float mb = sM[tid + s];
      if (mb > sM[tid]) {
        sM[tid] = mb; sC[tid] = sC[tid + s];
        sX[tid] = sX[tid + s]; sY[tid] = sY[tid + s]; sZ[tid] = sZ[tid + s];
      } else if (mb == sM[tid]) {
        sC[tid] += sC[tid + s];
        sX[tid] += sX[tid + s]; sY[tid] += sY[tid + s]; sZ[tid] += sZ[tid + s];
      }
    }
    __syncthreads();
  }

  if (tid == 0) {
    const float c = sC[0];
    pos[bj * 3 + 0] = sX[0] / c;
    pos[bj * 3 + 1] = sY[0] / c;
    pos[bj * 3 + 2] = sZ[0] / c;
  }
}

// ---------------------------------------------------------------------------
// Kernel 2: per-batch bbox -> world coords (max_r/min_r) + sample indices
//   idx[(b*3+a)*G + g] in [0, 63]
// ---------------------------------------------------------------------------
__global__ void k_bbox_idx(const float* __restrict__ pos,
                           const float* __restrict__ vmin,
                           const float* __restrict__ vmax,
                           const int*   __restrict__ boff,
                           float* __restrict__ out_maxr,
                           float* __restrict__ out_minr,
                           int*   __restrict__ idx) {
  const int b   = blockIdx.x;
  const int tid = threadIdx.x;                // 128 threads

  __shared__ float s_minr[3], s_scale[3], s_vmin[3], s_interval[3];

  if (tid == 0) {
    const float off = (float)boff[0];
#pragma unroll
    for (int a = 0; a < 3; ++a) {
      float mx = -INFINITY, mn = INFINITY;
      for (int j = 0; j < J; ++j) {
        const float v = pos[(b * J + j) * 3 + a];
        mx = fmaxf(mx, v);
        mn = fminf(mn, v);
      }
      const float maxb = fminf(fmaxf(mx + off, 0.f), (float)(D - 1));
      const float minb = fminf(fmaxf(mn - off, 0.f), (float)(D - 1));
      const float va = vmax[b * 3 + a], vi = vmin[b * 3 + a];
      const float range    = va - vi;
      const float interval = range / (float)(D - 1);
      const float maxr = vi + maxb / (float)(D - 1) * range;
      const float minr = vi + minb / (float)(D - 1) * range;
      out_maxr[b * 3 + a] = maxr;
      out_minr[b * 3 + a] = minr;
      s_minr[a] = minr; s_scale[a] = maxr - minr;
      s_vmin[a] = vi;   s_interval[a] = interval;
    }
  }
  __syncthreads();

  const float t = (float)tid / (float)(G - 1);     // linspace(0,1,G)
#pragma unroll
  for (int a = 0; a < 3; ++a) {
    const float axes = s_minr[a] + t * s_scale[a];
    const float na   = (axes - s_vmin[a]) / s_interval[a];
    const float r    = rintf(na);                  // round half to even (jnp.round)
    idx[(b * 3 + a) * G + tid] = (int)fminf(fmaxf(r, 0.f), (float)(D - 1));
  }
}

// ---------------------------------------------------------------------------
extern "C" void kernel_launch(void* const* d_in, const int* in_sizes, int n_in,
                              void* d_out, int out_size, void* d_ws, size_t ws_size,
                              hipStream_t stream) {
  const float* hm   = (const float*)d_in[0];
  // d_in[1] = vmin_s1 (unused by forward)
  const float* vmax = (const float*)d_in[2];
  const float* vmin = (const float*)d_in[3];
  const int*   boff = (const int*)d_in[4];

  float* out = (float*)d_out;
  const size_t interpN = (size_t)B * J * G3;   // 142,606,336
  float* out_maxr = out + interpN;             // [B,1,3]
  float* out_minr = out_maxr + B * 3;          // [B,1,3]

  float* pos = (float*)d_ws;                               // B*J*3 floats
  int*   idx = (int*)((char*)d_ws + 1024);                 // B*3*G ints

  k_argmax_mean<<<B * J, 256, 0, stream>>>(hm, pos);
  k_bbox_idx  <<<B,     128, 0, stream>>>(pos, vmin, vmax, boff,
                                          out_maxr, out_minr, idx);
  k_gather    <<<B * J * G, 256, 0, stream>>>(hm, idx, out);
}